// DecoderOutLayer_11476152615041
// MI455X (gfx1250) — compile-verified
//
#include <hip/hip_runtime.h>

typedef float v2f __attribute__((ext_vector_type(2)));
typedef float v8f __attribute__((ext_vector_type(8)));

#define TC      5
#define B_      4
#define C_      64
#define T_      1000
#define F_      96
#define O_      2
#define KTOT    (C_ * TC)            /* 320 */
#define NKC     (KTOT / 4)           /* 80 WMMA K-chunks of 4 */
#define TTILE   16
#define WIN     (TTILE + TC - 1)     /* 20 */
#define FCHUNK  8
#define TSTRIDE 68                   /* c-row stride: bank-conflict-free, even */
#define XSTRIDE_F 1364               /* per-f x-tile stride (20*68 + 4 pad)    */
#define WB_OFF  (FCHUNK * XSTRIDE_F)             /* weight table base (floats) */
#define WB_PER_F (KTOT * O_)                     /* 640 floats per frequency   */
#define LDS_FLOATS (WB_OFF + FCHUNK * WB_PER_F)  /* 16032 floats = 64128 B     */
#define NT_TILES   ((T_ + TTILE - 1) / TTILE)    /* 63 */
#define NF_CHUNKS  (F_ / FCHUNK)                 /* 12 */

// K remap: chunk kc in [0,80), kk in [0,4): c = (kc&15)*4 + kk, k = kc>>4.
// x tile stored [f][tl][c] and weights stored [f][kc][n][kk] so that each
// WMMA operand's two floats are contiguous + 8B-aligned -> one ds_load_b64
// per operand with a compile-time immediate offset (full unroll).
// Dead N-columns 2..15 (O=2) read clamped col-1 weights; never stored.

__global__ __launch_bounds__(256)
void decoder_out_wmma(const float* __restrict__ x,
                      const float* __restrict__ w,
                      const float* __restrict__ bias,
                      float* __restrict__ out) {
    __shared__ float lds[LDS_FLOATS];

    const int blk    = blockIdx.x;
    const int tTile  = blk % NT_TILES;
    const int fChunk = (blk / NT_TILES) % NF_CHUNKS;
    const int b      = blk / (NT_TILES * NF_CHUNKS);

    const int t0  = tTile * TTILE;
    const int f0  = fChunk * FCHUNK;
    const int tid = threadIdx.x;

    // ---- Stage x[b, :, t0-4 : t0+16, f0:f0+8] -> lds[fl][tl][c], f-coalesced ----
    for (int i = tid; i < C_ * WIN * FCHUNK; i += 256) {
        const int fl   = i & (FCHUNK - 1);
        const int pair = i >> 3;
        const int tl   = pair % WIN;
        const int c    = pair / WIN;
        const int tg   = t0 - (TC - 1) + tl;
        float v = 0.0f;
        if (tg >= 0 && tg < T_)
            v = x[((size_t)(b * C_ + c) * T_ + tg) * F_ + (f0 + fl)];
        lds[fl * XSTRIDE_F + tl * TSTRIDE + c] = v;
    }

    // ---- Stage weights into per-chunk B-fragment order:
    //      wB[fl][kc][n][kk] = w[f0+fl, c*5+k, n]   (5120 floats, LDS-linear) ----
    for (int d = tid; d < FCHUNK * WB_PER_F; d += 256) {
        const int fl = d / WB_PER_F;
        const int r  = d - fl * WB_PER_F;
        const int kc = r >> 3;
        const int n  = (r >> 2) & 1;
        const int kk = r & 3;
        const int c  = ((kc & 15) << 2) + kk;
        const int k  = kc >> 4;
        lds[WB_OFF + d] = w[(size_t)(f0 + fl) * WB_PER_F + (c * TC + k) * O_ + n];
    }
    __syncthreads();

    // ---- One wave per frequency: 80x v_wmma_f32_16x16x4_f32, all-LDS operands ----
    const int wave = tid >> 5;
    const int lane = tid & 31;
    const int half = lane >> 4;   // A: K-pair {0,1} vs {2,3}; D: M-half
    const int mn   = lane & 15;   // A: M row; B/D: N column
    const int f    = f0 + wave;

    // A base (even float index -> 8B aligned): + per-chunk const offset
    const float* lfA = &lds[wave * XSTRIDE_F + mn * TSTRIDE + half * 2];
    // B base: wB[wave][kc][n][kk], kk += half*2 folded, n clamped to {0,1}
    const int nclamp = (mn < O_) ? mn : 1;
    const float* lfB = &lds[WB_OFF + wave * WB_PER_F + nclamp * 4 + half * 2];

    v8f acc = {};

#pragma unroll
    for (int kc = 0; kc < NKC; ++kc) {
        const int aoff = (kc >> 4) * TSTRIDE + (kc & 15) * 4; // const, even
        const int boff = kc * 8;                              // const, even
        const v2f a  = *(const v2f*)(lfA + aoff);  // K = half*2 + {0,1}
        const v2f bf = *(const v2f*)(lfB + boff);  // same K pair, col n
        // (neg_a, A, neg_b, B, c_mod, C, reuse_a, reuse_b)
        acc = __builtin_amdgcn_wmma_f32_16x16x4_f32(
            false, a, false, bf, (short)0, acc, false, false);
    }

    // ---- D layout: VGPR r -> M = half*8 + r, N = mn; store the 2 real columns ----
    if (mn < O_) {
        const float bv = bias[f * O_ + mn];
#pragma unroll
        for (int r = 0; r < 8; ++r) {
            const int t = t0 + half * 8 + r;
            if (t < T_)
                out[((size_t)(b * O_ + mn) * T_ + t) * F_ + f] = acc[r] + bv;
        }
    }
}

extern "C" void kernel_launch(void* const* d_in, const int* in_sizes, int n_in,
                              void* d_out, int out_size, void* d_ws, size_t ws_size,
                              hipStream_t stream) {
    const float* x    = (const float*)d_in[0];  // [4,64,1000,96]
    const float* w    = (const float*)d_in[1];  // [96,320,2]
    const float* bias = (const float*)d_in[2];  // [96,2]
    float* out = (float*)d_out;                 // [4,2,1000,96]

    const int grid = B_ * NF_CHUNKS * NT_TILES; // 3024 workgroups
    decoder_out_wmma<<<grid, 256, 0, stream>>>(x, w, bias, out);
}